// GNN_36215164240659
// MI455X (gfx1250) — compile-verified
//
#include <hip/hip_runtime.h>

typedef __attribute__((ext_vector_type(2))) float v2f;
typedef __attribute__((ext_vector_type(8))) float v8f;

#define NFEAT 96   // IN_F == HID_F == 96 (K dim of both GEMMs)

// ---------------------------------------------------------------------------
// Zero a float region (re-run every launch for determinism).
// ---------------------------------------------------------------------------
__global__ void gcn_zero_kernel(float* __restrict__ p, int n) {
    int i = blockIdx.x * blockDim.x + threadIdx.x;
    if (i < n) p[i] = 0.0f;
}

// ---------------------------------------------------------------------------
// Per-edge degree accumulation: degOut[src]++, degIn[dst]++ (as floats).
// ---------------------------------------------------------------------------
__global__ void gcn_degree_kernel(const int* __restrict__ src,
                                  const int* __restrict__ dst,
                                  float* __restrict__ degOut,
                                  float* __restrict__ degIn, int nEdges) {
    int i = blockIdx.x * blockDim.x + threadIdx.x;
    if (i < nEdges) {
        atomicAdd(&degOut[src[i]], 1.0f);
        atomicAdd(&degIn[dst[i]], 1.0f);
    }
}

// ---------------------------------------------------------------------------
// In-place deg -> rsqrt(max(deg,1)) over both degree arrays (contiguous).
// ---------------------------------------------------------------------------
__global__ void gcn_norm_kernel(float* __restrict__ d, int n) {
    int i = blockIdx.x * blockDim.x + threadIdx.x;
    if (i < n) d[i] = rsqrtf(fmaxf(d[i], 1.0f));
}

// ---------------------------------------------------------------------------
// Edge scatter: agg[dst] += feat[src] * normOut[src].
// One wave per edge; 96 floats = 32 lanes * 3 consecutive floats (the wave
// reads one contiguous 384B row, then issues 3 global_atomic_add_f32/lane).
// All hot data (19.2 MB) lives in the 192MB L2, so this runs at L2 atomic
// throughput rather than HBM latency.
// ---------------------------------------------------------------------------
__global__ void gcn_edge_scatter_kernel(const float* __restrict__ feat,
                                        const float* __restrict__ normOut,
                                        const int* __restrict__ src,
                                        const int* __restrict__ dst,
                                        float* __restrict__ agg, int nEdges) {
    int gid  = blockIdx.x * blockDim.x + threadIdx.x;
    int edge = gid >> 5;
    int lane = gid & 31;
    if (edge >= nEdges) return;
    int s = src[edge];
    int d = dst[edge];
    float ns = normOut[s];
    const float* __restrict__ frow = feat + (size_t)s * NFEAT + lane * 3;
    float*       __restrict__ arow = agg  + (size_t)d * NFEAT + lane * 3;
    float v0 = frow[0] * ns;
    float v1 = frow[1] * ns;
    float v2 = frow[2] * ns;
    atomicAdd(arow + 0, v0);
    atomicAdd(arow + 1, v1);
    atomicAdd(arow + 2, v2);
}

// ---------------------------------------------------------------------------
// WMMA GEMM: out[M x NCOLS] = act( (A[M x 96] * normIn[row]) @ W + bias )
// NCOLS/RELU are compile-time so the accumulator array indexing is static and
// all 6 (or 2) v8f accumulators stay in VGPRs (no scratch spill).
//
// Block = 256 threads = 8 wave32s; each wave computes a 16-row strip over all
// NCOLS with V_WMMA_F32_16X16X4_F32. W (96xNCOLS f32) + bias staged in LDS.
//
// Fragment layouts per CDNA5 ISA (CDNA5 ISA):
//   A 16x4 f32:  lane m=L&15 is row M; v0 holds K = 2*(L>>4), v1 holds K+1.
//   B  4x16 f32: lane n=L&15 is col N; v0 holds K = 2*(L>>4), v1 holds K+1.
//   C/D 16x16:   vgpr r, lane L -> row = r + 8*(L>>4), col = L&15.
// EXEC stays all-1s: row guards are done by clamping loads / masking stores.
// ---------------------------------------------------------------------------
template <int NCOLS, bool RELU>
__global__ __launch_bounds__(256)
void gcn_gemm_wmma_kernel(const float* __restrict__ A,
                          const float* __restrict__ normIn,
                          const float* __restrict__ W,
                          const float* __restrict__ bias,
                          float* __restrict__ out, int M) {
    constexpr int NTILES = NCOLS / 16;   // 6 or 2

    __shared__ float sW[NFEAT * NCOLS];
    __shared__ float sB[NCOLS];

    const int tid = threadIdx.x;
    for (int i = tid; i < NFEAT * NCOLS; i += 256) sW[i] = W[i];
    if (tid < NCOLS) sB[tid] = bias[tid];
    __syncthreads();

    const int lane = tid & 31;
    const int wave = tid >> 5;
    const int m    = lane & 15;   // A row within tile / B-D column within tile
    const int half = lane >> 4;   // selects K pair (A/B) and row half (C/D)

    const int row0 = blockIdx.x * 128 + wave * 16;
    const int arow = min(row0 + m, M - 1);          // clamp for tail rows
    const float nin = normIn[arow];
    const float* __restrict__ Abase = A + (size_t)arow * NFEAT;

    v8f acc[NTILES] = {};

#pragma unroll
    for (int kk = 0; kk < NFEAT / 4; ++kk) {        // 24 K-steps of 4
        const int k = kk * 4 + half * 2;            // even -> 8B aligned
        v2f a = *(const v2f*)(Abase + k);
        a.x *= nin;                                 // fold D_in^{-1/2} scale
        a.y *= nin;
#pragma unroll
        for (int nt = 0; nt < NTILES; ++nt) {
            const int n = nt * 16 + m;
            v2f b;
            b.x = sW[k * NCOLS + n];
            b.y = sW[(k + 1) * NCOLS + n];
            acc[nt] = __builtin_amdgcn_wmma_f32_16x16x4_f32(
                false, a, false, b, (short)0, acc[nt], false, false);
        }
    }

#pragma unroll
    for (int nt = 0; nt < NTILES; ++nt) {
        const int col = nt * 16 + m;
        const float bv = sB[col];
#pragma unroll
        for (int r = 0; r < 8; ++r) {
            const int row = row0 + r + half * 8;
            float v = acc[nt][r] + bv;
            if (RELU) v = fmaxf(v, 0.0f);
            if (row < M) out[(size_t)row * NCOLS + col] = v;
        }
    }
}

// ---------------------------------------------------------------------------
// Host-side orchestration (graph-capture safe: kernels only, all on stream).
// ---------------------------------------------------------------------------
extern "C" void kernel_launch(void* const* d_in, const int* in_sizes, int n_in,
                              void* d_out, int out_size, void* d_ws, size_t ws_size,
                              hipStream_t stream) {
    const float* x   = (const float*)d_in[0];
    const int*   src = (const int*)d_in[1];
    const int*   dst = (const int*)d_in[2];
    const float* W1  = (const float*)d_in[3];
    const float* b1  = (const float*)d_in[4];
    const float* W2  = (const float*)d_in[5];
    const float* b2  = (const float*)d_in[6];
    float* out = (float*)d_out;

    const int N = in_sizes[0] / NFEAT;   // 50000 nodes
    const int E = in_sizes[1];           // 800000 edges

    float* ws      = (float*)d_ws;
    float* normOut = ws;                 // [N]   deg_out -> rsqrt norm
    float* normIn  = ws + N;             // [N]   deg_in  -> rsqrt norm
    float* agg     = ws + 2 * N;         // [N*96]
    float* h       = agg + (size_t)N * NFEAT;  // [N*96] hidden activations

    const int T = 256;
    const int degAggN = 2 * N + N * NFEAT;     // zero degs + agg together

    // --- degrees & norms ---------------------------------------------------
    gcn_zero_kernel<<<(degAggN + T - 1) / T, T, 0, stream>>>(ws, degAggN);
    gcn_degree_kernel<<<(E + T - 1) / T, T, 0, stream>>>(src, dst, normOut, normIn, E);
    gcn_norm_kernel<<<(2 * N + T - 1) / T, T, 0, stream>>>(ws, 2 * N);

    // --- layer 1: agg = sum_e x[src]*normOut[src]; h = relu(agg*nIn @ W1 + b1)
    {
        long long thr = (long long)E * 32;
        gcn_edge_scatter_kernel<<<(int)((thr + T - 1) / T), T, 0, stream>>>(
            x, normOut, src, dst, agg, E);
    }
    gcn_gemm_wmma_kernel<96, true><<<(N + 127) / 128, T, 0, stream>>>(
        agg, normIn, W1, b1, h, N);

    // --- layer 2 -----------------------------------------------------------
    gcn_zero_kernel<<<(N * NFEAT + T - 1) / T, T, 0, stream>>>(agg, N * NFEAT);
    {
        long long thr = (long long)E * 32;
        gcn_edge_scatter_kernel<<<(int)((thr + T - 1) / T), T, 0, stream>>>(
            h, normOut, src, dst, agg, E);
    }
    gcn_gemm_wmma_kernel<32, false><<<(N + 127) / 128, T, 0, stream>>>(
        agg, normIn, W2, b2, out, N);
}